// Loss_f_37452114821514
// MI455X (gfx1250) — compile-verified
//
#include <hip/hip_runtime.h>
#include <hip/hip_bf16.h>
#include <cfloat>

typedef __attribute__((ext_vector_type(2))) float v2f;
typedef __attribute__((ext_vector_type(8))) float v8f;

#define BATCH 2
#define ALPHA_F 0.3f

// ---------------------------------------------------------------------------
// Zero the two output scalars (harness poisons d_out with 0xAA).
// ---------------------------------------------------------------------------
__global__ void init_out_kernel(float* out) {
    if (threadIdx.x < 2) out[threadIdx.x] = 0.0f;
}

// ---------------------------------------------------------------------------
// Pack a point cloud into WMMA A-matrix and B-matrix lane layouts,
// interleaved per lane so a fragment is ONE global_load_b64 per lane:
//   tile block = 64 floats; entry (vgpr q, lane l) at offset l*2 + q.
//
// A (16x4 f32): lane l<16 = row M=l:   VGPR0=K0(x), VGPR1=K1(y)
//               lane l>=16 = row M=l-16: VGPR0=K2(z), VGPR1=K3(1)
// B (4x16 f32): VGPR0: lanes0-15 = K0 row, lanes16-31 = K2 row
//               VGPR1: lanes0-15 = K1 row, lanes16-31 = K3 row
//   A row n = (x0, x1, x2, 1);  B col m = (-2y0, -2y1, -2y2, |y|^2)
// => D[n,m] = |y|^2 - 2 x.y   (add |x|^2 after argmin; constant per row)
//
// Padded slots: A rows are guarded later; B pad cols get |y|^2 = 1e37 so
// they can never win the argmin.
// ---------------------------------------------------------------------------
__global__ void pack_points_kernel(const float* __restrict__ pts, int npts, int T,
                                   float* __restrict__ Apack, float* __restrict__ Bpack) {
    int tid = blockIdx.x * blockDim.x + threadIdx.x;
    int slots = T * 16;
    int total = BATCH * slots;
    if (tid >= total) return;
    int b = tid / slots;
    int s = tid - b * slots;
    int tile = s >> 4;
    int j = s & 15;

    float x = 0.0f, y = 0.0f, z = 0.0f, q = 1e37f, one = 0.0f;
    if (s < npts) {
        const float* p = pts + ((size_t)b * npts + s) * 3;
        x = p[0]; y = p[1]; z = p[2];
        q = x * x + y * y + z * z;
        one = 1.0f;
    }
    size_t base = ((size_t)b * T + tile) * 64;
    // A-matrix fragment: lane j -> (x, y) ; lane 16+j -> (z, 1)
    Apack[base + j * 2 + 0]        = x;
    Apack[base + j * 2 + 1]        = y;
    Apack[base + (16 + j) * 2 + 0] = z;
    Apack[base + (16 + j) * 2 + 1] = one;
    // B-matrix fragment: lane j -> (-2x, -2y) ; lane 16+j -> (-2z, |y|^2)
    Bpack[base + j * 2 + 0]        = -2.0f * x;
    Bpack[base + j * 2 + 1]        = -2.0f * y;
    Bpack[base + (16 + j) * 2 + 0] = -2.0f * z;
    Bpack[base + (16 + j) * 2 + 1] = q;
}

// ---------------------------------------------------------------------------
// One wave per (batch, x-tile). Loops over all y-tiles; each iteration is one
// global_load_b64 + one v_wmma_f32_16x16x4_f32 producing s = |y|^2 - 2 x.y
// for a 16x16 tile, then a running min/argmin per C slot (the serial VALU
// chain that sets throughput; unroll x4 lets loads/WMMAs hide under it).
// Finish with a 16-lane butterfly reduction, Welsch weighting + normal cosine
// loss, wave-sum, and one atomicAdd per wave.
// ---------------------------------------------------------------------------
__global__ void __launch_bounds__(32)
chamfer_dir_kernel(const float* __restrict__ srcPts, const float* __restrict__ srcNrm,
                   const float* __restrict__ dstNrm,
                   const float* __restrict__ Apack, const float* __restrict__ Bpack,
                   int nSrc, int nDst, int Tsrc, int Tdst,
                   float scale0, float scale1, float* __restrict__ out) {
    const int lane = threadIdx.x;
    const int b = blockIdx.x / Tsrc;
    const int xtile = blockIdx.x - b * Tsrc;

    // Load A fragment for this x-tile (constant across the whole loop).
    const float* Ab = Apack + ((size_t)b * Tsrc + xtile) * 64;
    v2f a = *(const v2f*)(Ab + lane * 2);

    float minv[8];
    int   mini[8];
#pragma unroll
    for (int r = 0; r < 8; ++r) { minv[r] = FLT_MAX; mini[r] = 0; }

    const float* Bb = Bpack + (size_t)b * Tdst * 64;
    v8f cz = {0.f, 0.f, 0.f, 0.f, 0.f, 0.f, 0.f, 0.f};

#pragma unroll 4
    for (int t = 0; t < Tdst; ++t) {
        v2f bb = *(const v2f*)(Bb + (size_t)t * 64 + lane * 2);
        // D = A x B + 0 : 16x16 tile of s = |y|^2 - 2 x.y
        v8f c = __builtin_amdgcn_wmma_f32_16x16x4_f32(
            /*neg_a=*/false, a, /*neg_b=*/false, bb,
            /*c_mod=*/(short)0, cz, /*reuse_a=*/false, /*reuse_b=*/false);
        const int yidx = (t << 4) + (lane & 15);
#pragma unroll
        for (int r = 0; r < 8; ++r) {
            float v = c[r];
            if (v < minv[r]) { minv[r] = v; mini[r] = yidx; }  // strict < keeps first idx
        }
    }

    // Butterfly min/argmin across the 16 lanes of each half-wave
    // (C slot r, lanes 0-15 => x-row r ; lanes 16-31 => x-row r+8).
#pragma unroll
    for (int r = 0; r < 8; ++r) {
#pragma unroll
        for (int off = 8; off >= 1; off >>= 1) {
            float ov = __shfl_xor(minv[r], off, 16);
            int   oi = __shfl_xor(mini[r], off, 16);
            if (ov < minv[r] || (ov == minv[r] && oi < mini[r])) {
                minv[r] = ov; mini[r] = oi;
            }
        }
    }

    const int sub = lane & 15;
    const int half = lane >> 4;
    // Lane `sub` (sub<8) in each half finalizes x-row = sub + 8*half.
    float myMin = minv[0];
    int   myIdx = mini[0];
#pragma unroll
    for (int r = 1; r < 8; ++r) {
        if (sub == r) { myMin = minv[r]; myIdx = mini[r]; }
    }

    float cham = 0.0f, nl = 0.0f;
    const int row = sub + half * 8;
    const int xi = (xtile << 4) + row;
    if (sub < 8 && xi < nSrc) {
        const size_t sOff = ((size_t)b * nSrc + xi) * 3;
        float px = srcPts[sOff + 0], py = srcPts[sOff + 1], pz = srcPts[sOff + 2];
        float dist = myMin + (px * px + py * py + pz * pz);   // full |x-y|^2
        dist = fmaxf(dist, 0.0f);
        const float inv2a2 = 1.0f / (2.0f * ALPHA_F * ALPHA_F);
        float w = expf(-(dist * dist) * inv2a2);               // Welsch on squared dist
        cham = w * dist * scale0;

        float ax = srcNrm[sOff + 0], ay = srcNrm[sOff + 1], az = srcNrm[sOff + 2];
        const size_t dOff = ((size_t)b * nDst + myIdx) * 3;
        float bx = dstNrm[dOff + 0], by = dstNrm[dOff + 1], bz = dstNrm[dOff + 2];
        float an = fmaxf(sqrtf(ax * ax + ay * ay + az * az), 1e-6f);
        float bn = fmaxf(sqrtf(bx * bx + by * by + bz * bz), 1e-6f);
        float cosv = (ax * bx + ay * by + az * bz) / (an * bn);
        nl = (1.0f - fabsf(cosv)) * scale1;
    }

    // Wave-wide sum (inactive lanes contribute 0), then one atomic per wave.
#pragma unroll
    for (int off = 16; off >= 1; off >>= 1) {
        cham += __shfl_xor(cham, off, 32);
        nl   += __shfl_xor(nl,   off, 32);
    }
    if (lane == 0) {
        atomicAdd(out + 0, cham);
        atomicAdd(out + 1, nl);
    }
}

// ---------------------------------------------------------------------------
// Launch: pack both clouds, then run both chamfer directions.
// Workspace: 4 packed arrays of BATCH*T*64 floats (~1.3 MB total).
// ---------------------------------------------------------------------------
extern "C" void kernel_launch(void* const* d_in, const int* in_sizes, int n_in,
                              void* d_out, int out_size, void* d_ws, size_t ws_size,
                              hipStream_t stream) {
    const float* predPts = (const float*)d_in[0];
    const float* predNrm = (const float*)d_in[1];
    const float* targPts = (const float*)d_in[2];
    const float* targNrm = (const float*)d_in[3];
    float* out = (float*)d_out;

    const int N = in_sizes[0] / (BATCH * 3);   // pred points per batch
    const int M = in_sizes[2] / (BATCH * 3);   // target points per batch
    const int Tp = (N + 15) / 16;
    const int Tt = (M + 15) / 16;

    float* ws = (float*)d_ws;
    float* Ap = ws;                              // A-pack, pred
    float* Bp = Ap + (size_t)BATCH * Tp * 64;    // B-pack, pred
    float* At = Bp + (size_t)BATCH * Tp * 64;    // A-pack, target
    float* Bt = At + (size_t)BATCH * Tt * 64;    // B-pack, target

    init_out_kernel<<<1, 32, 0, stream>>>(out);

    const int totP = BATCH * Tp * 16;
    const int totT = BATCH * Tt * 16;
    pack_points_kernel<<<(totP + 255) / 256, 256, 0, stream>>>(predPts, N, Tp, Ap, Bp);
    pack_points_kernel<<<(totT + 255) / 256, 256, 0, stream>>>(targPts, M, Tt, At, Bt);

    // Direction 1: target -> pred (source = target cloud)
    chamfer_dir_kernel<<<BATCH * Tt, 32, 0, stream>>>(
        targPts, targNrm, predNrm, At, Bp, M, N, Tt, Tp,
        1.0f / (float)BATCH, 1.0f / ((float)BATCH * (float)M), out);
    // Direction 2: pred -> target (source = pred cloud)
    chamfer_dir_kernel<<<BATCH * Tp, 32, 0, stream>>>(
        predPts, predNrm, targNrm, Ap, Bt, N, M, Tp, Tt,
        1.0f / (float)BATCH, 1.0f / ((float)BATCH * (float)N), out);
}